// InceptionBlock_84310208020812
// MI455X (gfx1250) — compile-verified
//
#include <hip/hip_runtime.h>

typedef __attribute__((ext_vector_type(2))) float v2f;
typedef __attribute__((ext_vector_type(8))) float v8f;

#define DIN 128
#define DOUT 128

static __device__ __forceinline__ v8f wmma4(v2f a, v2f b, v8f c) {
    return __builtin_amdgcn_wmma_f32_16x16x4_f32(
        /*neg_a=*/false, a, /*neg_b=*/false, b,
        /*c_mod=*/(short)0, c, /*reuse_a=*/false, /*reuse_b=*/false);
}

// ---------------------------------------------------------------------------
// Fused triple GEMM: reads X once, produces
//   Out0 = X @ Wln + bln   (straight into d_out)
//   H1   = X @ W1          (workspace, scatter source)
//   H2   = X @ W2          (workspace, scatter source)
// Block = 256 threads = 8 wave32s; block owns 32 M-rows (2 tiles of 16),
// wave w owns N-cols [16w,16w+16). 6 independent WMMA accumulator chains
// per wave; A fragments reused 3x, B fragments reused 2x.
// ---------------------------------------------------------------------------
__global__ __launch_bounds__(256) void fused_gemm3(
    const float* __restrict__ X,
    const float* __restrict__ Wln, const float* __restrict__ bln,
    const float* __restrict__ W1,  const float* __restrict__ W2,
    float* __restrict__ Out0, float* __restrict__ H1, float* __restrict__ H2,
    int M)
{
    const int wave = threadIdx.x >> 5;
    const int lane = threadIdx.x & 31;
    const int half = lane >> 4;        // 0: lanes 0-15, 1: lanes 16-31
    const int l    = lane & 15;
    const int m0   = blockIdx.x << 5;  // 32 rows per block
    const int n0   = wave << 4;
    if (m0 >= M) return;               // uniform per wave: EXEC all-ones

    // A fragments (16x4 fp32): lane<16 -> K={k,k+1}, lane>=16 -> K={k+2,k+3}
    const float* xrow0 = X + (size_t)(m0 + l) * DIN;
    const float* xrow1 = xrow0 + (size_t)16 * DIN;

    v8f c0a = {}, c0b = {};   // Wln, M-tile 0 / 1
    v8f c1a = {}, c1b = {};   // W1
    v8f c2a = {}, c2b = {};   // W2

#pragma unroll
    for (int k = 0; k < DIN; k += 4) {
        const int ka = k + (half << 1);
        v2f a0 = *(const v2f*)(xrow0 + ka);
        v2f a1 = *(const v2f*)(xrow1 + ka);

        // B fragments (4x16 fp32): rows ka, ka+1 of W, cols [n0, n0+16)
        const size_t bi = (size_t)ka * DOUT + n0 + l;
        v2f bL, b1, b2;
        bL.x = Wln[bi]; bL.y = Wln[bi + DOUT];
        b1.x = W1[bi];  b1.y = W1[bi + DOUT];
        b2.x = W2[bi];  b2.y = W2[bi + DOUT];

        c0a = wmma4(a0, bL, c0a);
        c0b = wmma4(a1, bL, c0b);
        c1a = wmma4(a0, b1, c1a);
        c1b = wmma4(a1, b1, c1b);
        c2a = wmma4(a0, b2, c2a);
        c2b = wmma4(a1, b2, c2b);
    }

    // C/D layout: VGPR v, lanes 0-15 -> M=v, N=l; lanes 16-31 -> M=8+v
    const int    ncol = n0 + l;
    const float  bv   = bln[ncol];
    const size_t r0   = (size_t)(m0 + (half << 3)) * DOUT + ncol;        // tile 0
    const size_t r1   = r0 + (size_t)16 * DOUT;                          // tile 1
#pragma unroll
    for (int v = 0; v < 8; ++v) {
        const size_t o = (size_t)v * DOUT;
        Out0[r0 + o] = c0a[v] + bv;
        Out0[r1 + o] = c0b[v] + bv;
        H1[r0 + o]   = c1a[v];
        H1[r1 + o]   = c1b[v];
        H2[r0 + o]   = c2a[v];
        H2[r1 + o]   = c2b[v];
    }
}

// ---------------------------------------------------------------------------
// Seed out1/out2 with their biases (scatter targets must start at bias).
// ---------------------------------------------------------------------------
__global__ __launch_bounds__(256) void init_bias(
    float* __restrict__ out1, const float* __restrict__ b1,
    float* __restrict__ out2, const float* __restrict__ b2, int total)
{
    int i = blockIdx.x * blockDim.x + threadIdx.x;
    if (i < total) {
        int c = i & (DOUT - 1);
        out1[i] = b1[c];
        out2[i] = b2[c];
    }
}

// ---------------------------------------------------------------------------
// Edge scatter: one wave per edge, lane handles 4 consecutive features.
// out[dst] += w_e * h[src] via hardware global_atomic_add_f32.
// ---------------------------------------------------------------------------
__global__ __launch_bounds__(256) void scatter_edges(
    const float* __restrict__ H, const int* __restrict__ src,
    const int* __restrict__ dst, const float* __restrict__ ew,
    float* __restrict__ Out, int E)
{
    const int waveId = blockIdx.x * (blockDim.x >> 5) + (threadIdx.x >> 5);
    const int lane   = threadIdx.x & 31;
    if (waveId >= E) return;

    const int   s  = src[waveId];
    const int   d  = dst[waveId];
    const float we = ew[waveId];

    const float4 hv = *(const float4*)(H + (size_t)s * DOUT + lane * 4);
    float* op = Out + (size_t)d * DOUT + lane * 4;
#if defined(__HIP_DEVICE_COMPILE__)
    unsafeAtomicAdd(op + 0, hv.x * we);
    unsafeAtomicAdd(op + 1, hv.y * we);
    unsafeAtomicAdd(op + 2, hv.z * we);
    unsafeAtomicAdd(op + 3, hv.w * we);
#endif
}

// ---------------------------------------------------------------------------
extern "C" void kernel_launch(void* const* d_in, const int* in_sizes, int n_in,
                              void* d_out, int out_size, void* d_ws, size_t ws_size,
                              hipStream_t stream)
{
    const float* x   = (const float*)d_in[0];
    const int*   ei1 = (const int*)  d_in[1];   // [2,E]: row0=src, row1=dst
    const float* ew1 = (const float*)d_in[2];
    const int*   ei2 = (const int*)  d_in[3];
    const float* ew2 = (const float*)d_in[4];
    const float* Wln = (const float*)d_in[5];
    const float* bln = (const float*)d_in[6];
    const float* W1  = (const float*)d_in[7];
    const float* b1  = (const float*)d_in[8];
    const float* W2  = (const float*)d_in[9];
    const float* b2  = (const float*)d_in[10];

    const int M  = in_sizes[0] / DIN;   // 100000
    const int E1 = in_sizes[2];         // 600000
    const int E2 = in_sizes[4];

    float* out0 = (float*)d_out;
    float* out1 = out0 + (size_t)M * DOUT;
    float* out2 = out1 + (size_t)M * DOUT;
    float* h1   = (float*)d_ws;
    float* h2   = h1 + (size_t)M * DOUT;

    // 1) seed scatter targets with bias
    {
        const int total   = M * DOUT;
        const int threads = 256;
        const int blocks  = (total + threads - 1) / threads;
        init_bias<<<blocks, threads, 0, stream>>>(out1, b1, out2, b2, total);
    }

    // 2) fused triple WMMA GEMM (x read once; 6 independent wmma chains/wave)
    {
        const int blocks = (M + 31) / 32;
        fused_gemm3<<<blocks, 256, 0, stream>>>(x, Wln, bln, W1, W2,
                                                out0, h1, h2, M);
    }

    // 3) edge scatters (gathers hit L2: each h is 51.2MB < 192MB L2)
    {
        const int wavesPerBlock = 8;    // 256 / 32
        const int blocks1 = (E1 + wavesPerBlock - 1) / wavesPerBlock;
        scatter_edges<<<blocks1, 256, 0, stream>>>(h1, ei1, ei1 + E1, ew1, out1, E1);
        const int blocks2 = (E2 + wavesPerBlock - 1) / wavesPerBlock;
        scatter_edges<<<blocks2, 256, 0, stream>>>(h2, ei2, ei2 + E2, ew2, out2, E2);
    }
}